// EuclideanAttentionBlock_61546881352455
// MI455X (gfx1250) — compile-verified
//
#include <hip/hip_runtime.h>
#include <math.h>

typedef __attribute__((ext_vector_type(2))) float v2f;
typedef __attribute__((ext_vector_type(8))) float v8f;

#define N_NODES 32768
#define E_EDGES 524288
#define F_DIM   128
#define NRBF    32
#define EV_DIM  16
#define H_HEADS 4
#define D_HEAD  32
#define INV_SQRT_D 0.17677669529663687f   // 1/sqrt(32)

// packed B-fragment array sizes (in v2f elements)
#define NODE_PACK_ELEMS (8 * 8 * 32)      // [wave(8)][k(8)][lane(32)]
#define EDGE_PACK_ELEMS (8 * 9 * 32)      // [ct(8)][k(9)][lane(32)]

__device__ __forceinline__ float silu(float x) {
    return x / (1.0f + __expf(-x));
}

// ---------------------------------------------------------------------------
// Pack node-projection weights [4,32,32] into WMMA-B fragment-major layout:
//   pack[((w*8)+k)*32 + lane] = { W[h][kk][colN], W[h][kk+1][colN] }
//   w = h*2+half, kk = 4k + 2*(lane>>4), colN = half*16 + (lane&15)
// One coalesced b64 load per fragment afterwards.
// ---------------------------------------------------------------------------
__global__ void pack_node_weights(const float* __restrict__ W,  // one [4,32,32]
                                  v2f* __restrict__ pack)
{
    int i = blockIdx.x * blockDim.x + threadIdx.x;
    if (i >= NODE_PACK_ELEMS) return;
    const int lane = i & 31;
    const int k    = (i >> 5) & 7;
    const int w    = i >> 8;            // 0..7
    const int h    = w >> 1;
    const int half = w & 1;
    const int hi   = lane >> 4;
    const int row  = lane & 15;
    const int kk   = 4 * k + 2 * hi;
    const int colN = half * 16 + row;
    v2f b;
    b.x = W[h * (D_HEAD * D_HEAD) + kk * D_HEAD + colN];
    b.y = W[h * (D_HEAD * D_HEAD) + (kk + 1) * D_HEAD + colN];
    pack[i] = b;
}

// ---------------------------------------------------------------------------
// Pack edge-filter weights [36,128] into fragment-major layout:
//   pack[((ct*9)+k)*32 + lane] = { Wf[kk][col], Wf[kk+1][col] }
//   kk = 4k + 2*(lane>>4), col = ct*16 + (lane&15)
// ---------------------------------------------------------------------------
__global__ void pack_edge_weights(const float* __restrict__ Wf, // [36,128]
                                  v2f* __restrict__ pack)
{
    int i = blockIdx.x * blockDim.x + threadIdx.x;
    if (i >= EDGE_PACK_ELEMS) return;
    const int lane = i & 31;
    const int k    = (i >> 5) % 9;
    const int ct   = i / (9 * 32);
    const int hi   = lane >> 4;
    const int row  = lane & 15;
    const int kk   = 4 * k + 2 * hi;
    const int col  = ct * 16 + row;
    v2f b;
    b.x = Wf[kk * F_DIM + col];
    b.y = Wf[(kk + 1) * F_DIM + col];
    pack[i] = b;
}

// ---------------------------------------------------------------------------
// Kernel 1: node projections via V_WMMA_F32_16X16X4_F32.
// Block = 256 threads = 8 waves; tile of 16 nodes in LDS.
// Wave w: head = w>>1, output half = w&1. 8 K-steps x 5 weight mats = 40 WMMA.
// B fragments: one coalesced global_load_b64 per matrix per K-step.
// ---------------------------------------------------------------------------
__global__ __launch_bounds__(256) void node_proj_kernel(
    const float* __restrict__ x,        // [N,128]
    const v2f* __restrict__ Pq,         // packed fragments
    const v2f* __restrict__ Pk,
    const v2f* __restrict__ Pv,
    const v2f* __restrict__ Pqe,
    const v2f* __restrict__ Pke,
    float* __restrict__ q_inv, float* __restrict__ k_inv,
    float* __restrict__ v_inv,
    float* __restrict__ q_ev,  float* __restrict__ k_ev)
{
    __shared__ float xt[16 * F_DIM];
    const int t = threadIdx.x;
    const int node0 = blockIdx.x * 16;

    #pragma unroll
    for (int i = 0; i < 8; ++i)
        xt[t + i * 256] = x[(size_t)node0 * F_DIM + t + i * 256];
    __syncthreads();

    const int wave = t >> 5;
    const int lane = t & 31;
    const int h    = wave >> 1;        // head 0..3
    const int half = wave & 1;         // which 16-col output half
    const int row  = lane & 15;        // A-frag M index / D-frag N index
    const int hi   = lane >> 4;        // lane half: selects K pair, M offset
    const int colN = half * 16 + row;  // output col inside head (0..31)

    v8f aq  = {0,0,0,0,0,0,0,0};
    v8f ak  = aq, av = aq, aqe = aq, ake = aq;

    const int pbase = wave * 8 * 32 + lane;

    #pragma unroll
    for (int k = 0; k < 8; ++k) {
        const int kk = 4 * k + 2 * hi;  // this lane-half's K pair
        v2f a = *(const v2f*)&xt[row * F_DIM + h * D_HEAD + kk];
        const int pi = pbase + k * 32;
        aq  = __builtin_amdgcn_wmma_f32_16x16x4_f32(false, a, false, Pq[pi],  (short)0, aq,  false, false);
        ak  = __builtin_amdgcn_wmma_f32_16x16x4_f32(false, a, false, Pk[pi],  (short)0, ak,  false, false);
        av  = __builtin_amdgcn_wmma_f32_16x16x4_f32(false, a, false, Pv[pi],  (short)0, av,  false, false);
        aqe = __builtin_amdgcn_wmma_f32_16x16x4_f32(false, a, false, Pqe[pi], (short)0, aqe, false, false);
        ake = __builtin_amdgcn_wmma_f32_16x16x4_f32(false, a, false, Pke[pi], (short)0, ake, false, false);
    }

    const int colG = h * D_HEAD + colN;
    #pragma unroll
    for (int r = 0; r < 8; ++r) {
        const int m = r + hi * 8;               // D layout: lanes 16-31 -> M+8
        const size_t o = (size_t)(node0 + m) * F_DIM + colG;
        q_inv[o] = silu(aq[r]);
        k_inv[o] = silu(ak[r]);
        v_inv[o] = av[r];
        q_ev[o]  = silu(aqe[r]);
        k_ev[o]  = silu(ake[r]);
    }
}

// ---------------------------------------------------------------------------
// Kernel 2: zero the (harness-poisoned) output accumulator.
// ---------------------------------------------------------------------------
__global__ void zero_kernel(float* __restrict__ p, int n) {
    int i = blockIdx.x * blockDim.x + threadIdx.x;
    if (i < n) p[i] = 0.0f;
}

// ---------------------------------------------------------------------------
// Kernel 3: per-edge filter GEMM (WMMA) + attention + atomic scatter.
// Block = 64 threads (2 waves), 32 edges, ~38 KB LDS.
// ---------------------------------------------------------------------------
__global__ __launch_bounds__(64) void edge_kernel(
    const float* __restrict__ ev_features, // [N,16]
    const float* __restrict__ rbf,         // [E,32]
    const float* __restrict__ sh,          // [E,16]
    const float* __restrict__ cutoffs,     // [E]
    const v2f*   __restrict__ PfI,         // packed Wf_inv fragments
    const float* __restrict__ bf_inv,      // [128]
    const v2f*   __restrict__ PfE,         // packed Wf_ev fragments
    const float* __restrict__ bf_ev,       // [128]
    const int*   __restrict__ senders,
    const int*   __restrict__ receivers,
    const float* __restrict__ q_inv, const float* __restrict__ k_inv,
    const float* __restrict__ v_inv,
    const float* __restrict__ q_ev,  const float* __restrict__ k_ev,
    float* __restrict__ d_inv,             // [N,128]
    float* __restrict__ d_ev)              // [N,16]
{
    __shared__ float feat[32][40];         // cols 0..31 rbf, 32..35 invariants
    __shared__ float fwI[32][F_DIM];
    __shared__ float fwE[32][F_DIM];
    __shared__ float cut[32];
    __shared__ int   sN[32], rN[32];

    const int t  = threadIdx.x;
    const int e0 = blockIdx.x * 32;

    // ---- stage 1: build feat rows (2 threads per edge) ----
    {
        const int e  = t >> 1;
        const int ge = e0 + e;
        if ((t & 1) == 0) {
            const int s = senders[ge];
            const int r = receivers[ge];
            sN[e] = s; rN[e] = r;
            cut[e] = cutoffs[ge];
            float i0 = 0.f, i1 = 0.f, i2 = 0.f, i3 = 0.f;
            const float* es = ev_features + (size_t)s * EV_DIM;
            const float* er = ev_features + (size_t)r * EV_DIM;
            #pragma unroll
            for (int j = 0; j < EV_DIM; ++j) {
                float df = es[j] - er[j];
                float sq = df * df;
                if (j == 0)      i0 += sq;
                else if (j < 4)  i1 += sq;
                else if (j < 9)  i2 += sq;
                else             i3 += sq;
            }
            feat[e][32] = i0; feat[e][33] = i1;
            feat[e][34] = i2; feat[e][35] = i3;
        } else {
            const float4* rb = (const float4*)(rbf + (size_t)ge * NRBF);
            #pragma unroll
            for (int j = 0; j < 8; ++j) {
                float4 vv = rb[j];
                feat[e][4*j+0] = vv.x; feat[e][4*j+1] = vv.y;
                feat[e][4*j+2] = vv.z; feat[e][4*j+3] = vv.w;
            }
        }
    }
    __syncthreads();

    // ---- stage 2: fw = feat @ Wf (+bias) via WMMA f32 16x16x4, K=36=9x4 ----
    {
        const int wave = t >> 5;
        const int lane = t & 31;
        const int row  = lane & 15;
        const int hi   = lane >> 4;
        const int er   = wave * 16 + row;          // A-frag edge row

        for (int ct = 0; ct < 8; ++ct) {           // 8 column tiles of 16
            v8f accI = {0,0,0,0,0,0,0,0};
            v8f accE = accI;
            const int col = ct * 16 + row;
            const int pb  = ct * 9 * 32 + lane;
            #pragma unroll
            for (int k = 0; k < 9; ++k) {          // k=8 covers invariant rows
                const int kk = 4 * k + 2 * hi;
                v2f a = *(const v2f*)&feat[er][kk];
                const int pi = pb + k * 32;
                accI = __builtin_amdgcn_wmma_f32_16x16x4_f32(false, a, false, PfI[pi], (short)0, accI, false, false);
                accE = __builtin_amdgcn_wmma_f32_16x16x4_f32(false, a, false, PfE[pi], (short)0, accE, false, false);
            }
            const float bi = bf_inv[col];
            const float be = bf_ev[col];
            #pragma unroll
            for (int r = 0; r < 8; ++r) {
                const int m = wave * 16 + r + hi * 8;
                fwI[m][col] = accI[r] + bi;
                fwE[m][col] = accE[r] + be;
            }
        }
    }
    __syncthreads();

    // ---- stage 3: attention + atomic scatter, one (edge, head) per item ----
    #pragma unroll
    for (int rep = 0; rep < 2; ++rep) {
        const int pair = rep * 64 + t;     // 0..127 : 32 edges x 4 heads
        const int e  = pair >> 2;
        const int h  = pair & 3;
        const int ge = e0 + e;
        const int s  = sN[e];
        const int r  = rN[e];
        const float c = cut[e];
        const size_t bs = (size_t)s * F_DIM + h * D_HEAD;
        const size_t br = (size_t)r * F_DIM + h * D_HEAD;

        float aI = 0.f, aE = 0.f;
        #pragma unroll
        for (int d4 = 0; d4 < 8; ++d4) {
            float4 qi = *(const float4*)(q_inv + br + 4*d4);
            float4 ki = *(const float4*)(k_inv + bs + 4*d4);
            const float* fI = &fwI[e][h*D_HEAD + 4*d4];
            aI += qi.x*ki.x*fI[0] + qi.y*ki.y*fI[1] + qi.z*ki.z*fI[2] + qi.w*ki.w*fI[3];
            float4 qe = *(const float4*)(q_ev + br + 4*d4);
            float4 ke = *(const float4*)(k_ev + bs + 4*d4);
            const float* fE = &fwE[e][h*D_HEAD + 4*d4];
            aE += qe.x*ke.x*fE[0] + qe.y*ke.y*fE[1] + qe.z*ke.z*fE[2] + qe.w*ke.w*fE[3];
        }
        const float scI = aI * INV_SQRT_D * c;
        const float scE = aE * INV_SQRT_D * c;

        float* dI = d_inv + br;
        #pragma unroll
        for (int d4 = 0; d4 < 8; ++d4) {
            float4 vv = *(const float4*)(v_inv + bs + 4*d4);
            atomicAdd(dI + 4*d4 + 0, scI * vv.x);
            atomicAdd(dI + 4*d4 + 1, scI * vv.y);
            atomicAdd(dI + 4*d4 + 2, scI * vv.z);
            atomicAdd(dI + 4*d4 + 3, scI * vv.w);
        }

        // ev messages: degree h covers indices off[h] .. off[h]+2h
        const int off  = (h == 0) ? 0 : (h == 1) ? 1 : (h == 2) ? 4 : 9;
        const int nrep = 2 * h + 1;
        const float* shp = sh + (size_t)ge * EV_DIM;
        float* dE = d_ev + (size_t)r * EV_DIM;
        for (int j = 0; j < nrep; ++j)
            atomicAdd(dE + off + j, scE * shp[off + j]);
    }
}

// ---------------------------------------------------------------------------
extern "C" void kernel_launch(void* const* d_in, const int* in_sizes, int n_in,
                              void* d_out, int out_size, void* d_ws, size_t ws_size,
                              hipStream_t stream) {
    const float* inv_features = (const float*)d_in[0];
    const float* ev_features  = (const float*)d_in[1];
    const float* rbf          = (const float*)d_in[2];
    const float* sh_vectors   = (const float*)d_in[3];
    const float* cutoffs      = (const float*)d_in[4];
    const float* W_q_inv      = (const float*)d_in[5];
    const float* W_k_inv      = (const float*)d_in[6];
    const float* W_v_inv      = (const float*)d_in[7];
    const float* W_q_ev       = (const float*)d_in[8];
    const float* W_k_ev       = (const float*)d_in[9];
    const float* Wf_inv       = (const float*)d_in[10];
    const float* bf_inv       = (const float*)d_in[11];
    const float* Wf_ev        = (const float*)d_in[12];
    const float* bf_ev        = (const float*)d_in[13];
    const int*   senders      = (const int*)d_in[14];
    const int*   receivers    = (const int*)d_in[15];

    float* out   = (float*)d_out;
    float* d_inv = out;                                   // [N,128]
    float* d_ev  = out + (size_t)N_NODES * F_DIM;         // [N,16]

    const size_t NF = (size_t)N_NODES * F_DIM;
    float* q_inv = (float*)d_ws;
    float* k_inv = q_inv + NF;
    float* v_inv = k_inv + NF;
    float* q_ev  = v_inv + NF;
    float* k_ev  = q_ev  + NF;
    v2f* packBase = (v2f*)(k_ev + NF);
    v2f* Pq  = packBase;
    v2f* Pk  = Pq  + NODE_PACK_ELEMS;
    v2f* Pv  = Pk  + NODE_PACK_ELEMS;
    v2f* Pqe = Pv  + NODE_PACK_ELEMS;
    v2f* Pke = Pqe + NODE_PACK_ELEMS;
    v2f* PfI = Pke + NODE_PACK_ELEMS;
    v2f* PfE = PfI + EDGE_PACK_ELEMS;

    // one-shot weight repacking into WMMA-fragment-major layout
    pack_node_weights<<<(NODE_PACK_ELEMS + 255) / 256, 256, 0, stream>>>(W_q_inv, Pq);
    pack_node_weights<<<(NODE_PACK_ELEMS + 255) / 256, 256, 0, stream>>>(W_k_inv, Pk);
    pack_node_weights<<<(NODE_PACK_ELEMS + 255) / 256, 256, 0, stream>>>(W_v_inv, Pv);
    pack_node_weights<<<(NODE_PACK_ELEMS + 255) / 256, 256, 0, stream>>>(W_q_ev,  Pqe);
    pack_node_weights<<<(NODE_PACK_ELEMS + 255) / 256, 256, 0, stream>>>(W_k_ev,  Pke);
    pack_edge_weights<<<(EDGE_PACK_ELEMS + 255) / 256, 256, 0, stream>>>(Wf_inv, PfI);
    pack_edge_weights<<<(EDGE_PACK_ELEMS + 255) / 256, 256, 0, stream>>>(Wf_ev,  PfE);

    node_proj_kernel<<<N_NODES / 16, 256, 0, stream>>>(
        inv_features, Pq, Pk, Pv, Pqe, Pke,
        q_inv, k_inv, v_inv, q_ev, k_ev);

    zero_kernel<<<(out_size + 255) / 256, 256, 0, stream>>>(out, out_size);

    edge_kernel<<<E_EDGES / 32, 64, 0, stream>>>(
        ev_features, rbf, sh_vectors, cutoffs,
        PfI, bf_inv, PfE, bf_ev, senders, receivers,
        q_inv, k_inv, v_inv, q_ev, k_ev, d_inv, d_ev);
}